// SkipGram_23922967839117
// MI455X (gfx1250) — compile-verified
//
#include <hip/hip_runtime.h>
#include <math.h>

// CDNA5 / gfx1250: wave32, WMMA f32 16x16x4 for the f32 GEMM.
typedef __attribute__((ext_vector_type(2))) float v2f;
typedef __attribute__((ext_vector_type(8))) float v8f;

constexpr int KH      = 128;           // hidden dim == GEMM K (fixed by model)
constexpr int WAVE_N  = 64;            // output cols per wave (4 x 16-col WMMA tiles)
constexpr int BLOCK_N = 8 * WAVE_N;    // 512 cols per 256-thread (8-wave) block
constexpr int BLOCK_M = 32;           // rows per block: 2 x 16-row WMMA subtiles/wave

// ---------------------------------------------------------------------------
// Kernel 1: hidden[b][h] = W_enc[h][idx[b]] + b_enc[h]   (column gather)
// ---------------------------------------------------------------------------
__global__ void sg_gather_hidden(const int* __restrict__ idx,
                                 const float* __restrict__ W_enc,
                                 const float* __restrict__ b_enc,
                                 float* __restrict__ hidden,
                                 int B, int H, int V) {
    int t = blockIdx.x * blockDim.x + threadIdx.x;
    if (t >= B * H) return;
    int b = t / H;
    int h = t - b * H;
    hidden[t] = W_enc[(size_t)h * (size_t)V + (size_t)idx[b]] + b_enc[h];
}

// ---------------------------------------------------------------------------
// Kernel 2: logits[b][v] = dot(hidden[b,:], W_dec[v,:]) + b_dec[v]
// NT GEMM, K = 128 contiguous for both operands.
//
// Block (256 thr, 8 waves): 32 rows x 512 cols. A tile (32x128, 16 KB) is
// staged in LDS once and shared by all 8 waves (ds_load_b64 fragments).
// Each wave: 32(M) x 64(N) -> 8 v8f accumulators, B fragments double-buffered.
//
// ISA fragment layout (32-bit A 16x4): lanes 0-15 hold M=lane, K={k,k+1};
// lanes 16-31 hold M=lane-16, K={k+2,k+3}. B (4x16) mirrors with N=lane&15.
// C/D v8f: component r -> row = r + (lane>>4)*8, col = lane&15.
// ---------------------------------------------------------------------------
__global__ void sg_wmma_logits(const float* __restrict__ hidden,
                               const float* __restrict__ W_dec,
                               const float* __restrict__ b_dec,
                               float* __restrict__ out,
                               int B, int V) {
    __shared__ float ldsA[BLOCK_M * KH];   // 16 KB

    const int tid  = threadIdx.x;
    const int lane = tid & 31;
    const int wave = tid >> 5;
    const int lh   = lane & 15;   // M (A) / N (B) within 16-wide tile
    const int lg   = lane >> 4;   // K half-select

    const int row0 = blockIdx.y * BLOCK_M;
    const int v0   = blockIdx.x * BLOCK_N + wave * WAVE_N;

    // ---- cooperative stage of the 32x128 A tile into LDS (float4) ----
    {
        const float4* h4 = (const float4*)hidden;
        float4* l4 = (float4*)ldsA;
        const int C4 = KH >> 2;                      // 32 float4 per row
#pragma unroll
        for (int t = tid; t < BLOCK_M * C4; t += 256) {
            int r  = t >> 5;                          // t / C4
            int c  = t & (C4 - 1);
            int gr = row0 + r;
            if (gr >= B) gr = B - 1;                  // clamp (stores guarded)
            l4[t] = h4[(size_t)gr * C4 + c];
        }
    }
    __syncthreads();

    if (v0 >= V) return;          // wave-uniform exit (no further barriers)

    // ---- B fragment bases for the 4 column sub-tiles (clamped: WMMA needs
    //      full EXEC, so OOB lanes read a valid dummy row; stores guarded) ----
    const float* bptr[4];
    int vcol[4];
#pragma unroll
    for (int j = 0; j < 4; ++j) {
        int v = v0 + j * 16 + lh;
        vcol[j] = v;
        int vc = (v < V) ? v : (V - 1);
        bptr[j] = W_dec + (size_t)vc * KH + (size_t)(lg * 2);
        __builtin_prefetch(bptr[j], 0, 0);            // global_prefetch_b8
        __builtin_prefetch(bptr[j] + 64, 0, 0);
    }

    // A fragment bases in LDS for the two 16-row subtiles
    const float* aBase0 = ldsA + (lh)      * KH + lg * 2;
    const float* aBase1 = ldsA + (16 + lh) * KH + lg * 2;

    v8f acc0[4], acc1[4];
#pragma unroll
    for (int j = 0; j < 4; ++j) {
        acc0[j] = (v8f){0.f, 0.f, 0.f, 0.f, 0.f, 0.f, 0.f, 0.f};
        acc1[j] = (v8f){0.f, 0.f, 0.f, 0.f, 0.f, 0.f, 0.f, 0.f};
    }

    // ---- K loop, B double-buffered so WMMAs overlap the next loads ----
    v2f bc[4];
#pragma unroll
    for (int j = 0; j < 4; ++j) bc[j] = *(const v2f*)(bptr[j]);

#pragma unroll 4
    for (int k = 0; k < KH; k += 4) {
        const int kn = (k + 4 < KH) ? (k + 4) : 0;    // last iter: dummy reload
        v2f bn[4];
#pragma unroll
        for (int j = 0; j < 4; ++j) bn[j] = *(const v2f*)(bptr[j] + kn);

        v2f a0 = *(const v2f*)(aBase0 + k);           // ds_load_b64
        v2f a1 = *(const v2f*)(aBase1 + k);

#pragma unroll
        for (int j = 0; j < 4; ++j) {
            // 8 args: (neg_a, A, neg_b, B, c_mod, C, reuse_a, reuse_b)
            acc0[j] = __builtin_amdgcn_wmma_f32_16x16x4_f32(
                false, a0, false, bc[j], (short)0, acc0[j], false, false);
            acc1[j] = __builtin_amdgcn_wmma_f32_16x16x4_f32(
                false, a1, false, bc[j], (short)0, acc1[j], false, false);
        }
#pragma unroll
        for (int j = 0; j < 4; ++j) bc[j] = bn[j];
    }

    // ---- bias + guarded stores ----
#pragma unroll
    for (int j = 0; j < 4; ++j) {
        int v = vcol[j];
        if (v < V) {
            float bias = b_dec[v];
#pragma unroll
            for (int r = 0; r < 8; ++r) {
                int rowa = row0 + lg * 8 + r;
                int rowb = rowa + 16;
                if (rowa < B) out[(size_t)rowa * (size_t)V + (size_t)v] = acc0[j][r] + bias;
                if (rowb < B) out[(size_t)rowb * (size_t)V + (size_t)v] = acc1[j][r] + bias;
            }
        }
    }
}

// ---------------------------------------------------------------------------
// Kernel 3: in-place row-wise log_softmax over V elements.
// One 256-thread (8-wave) block per row. wave32 shuffles + LDS cross-wave.
// ---------------------------------------------------------------------------
__device__ __forceinline__ float sg_wave_max(float v) {
#pragma unroll
    for (int off = 16; off > 0; off >>= 1)
        v = fmaxf(v, __shfl_down(v, off, 32));
    return v;
}

__device__ __forceinline__ float sg_wave_sum(float v) {
#pragma unroll
    for (int off = 16; off > 0; off >>= 1)
        v += __shfl_down(v, off, 32);
    return v;
}

__global__ void sg_logsoftmax_rows(float* __restrict__ out, int V) {
    __shared__ float red[8];
    __shared__ float bcast[2];

    const int tid  = threadIdx.x;
    const int lane = tid & 31;
    const int wv   = tid >> 5;
    float* row = out + (size_t)blockIdx.x * (size_t)V;
    const int n4 = V >> 2;
    const float4* row4 = (const float4*)row;

    // ---- phase 1: row max ----
    float m = -INFINITY;
    for (int i = tid; i < n4; i += blockDim.x) {
        float4 x = row4[i];
        m = fmaxf(m, fmaxf(fmaxf(x.x, x.y), fmaxf(x.z, x.w)));
    }
    for (int i = (n4 << 2) + tid; i < V; i += blockDim.x) m = fmaxf(m, row[i]);
    m = sg_wave_max(m);
    if (lane == 0) red[wv] = m;
    __syncthreads();
    if (tid == 0) {
        float x = red[0];
#pragma unroll
        for (int i = 1; i < 8; ++i) x = fmaxf(x, red[i]);
        bcast[0] = x;
    }
    __syncthreads();
    const float M = bcast[0];
    __syncthreads();   // red[] reuse hazard

    // ---- phase 2: sum of exp(x - M) ----
    float s = 0.f;
    for (int i = tid; i < n4; i += blockDim.x) {
        float4 x = row4[i];
        s += expf(x.x - M) + expf(x.y - M) + expf(x.z - M) + expf(x.w - M);
    }
    for (int i = (n4 << 2) + tid; i < V; i += blockDim.x) s += expf(row[i] - M);
    s = sg_wave_sum(s);
    if (lane == 0) red[wv] = s;
    __syncthreads();
    if (tid == 0) {
        float x = 0.f;
#pragma unroll
        for (int i = 0; i < 8; ++i) x += red[i];
        bcast[1] = M + logf(x);
    }
    __syncthreads();
    const float corr = bcast[1];

    // ---- phase 3: write log_softmax = x - (M + log S) ----
    float4* row4w = (float4*)row;
    for (int i = tid; i < n4; i += blockDim.x) {
        float4 x = row4w[i];
        x.x -= corr; x.y -= corr; x.z -= corr; x.w -= corr;
        row4w[i] = x;
    }
    for (int i = (n4 << 2) + tid; i < V; i += blockDim.x) row[i] -= corr;
}

// ---------------------------------------------------------------------------
// Launch
// ---------------------------------------------------------------------------
extern "C" void kernel_launch(void* const* d_in, const int* in_sizes, int n_in,
                              void* d_out, int out_size, void* d_ws, size_t ws_size,
                              hipStream_t stream) {
    const int*   idx   = (const int*)d_in[0];
    const float* W_enc = (const float*)d_in[1];
    const float* b_enc = (const float*)d_in[2];
    const float* W_dec = (const float*)d_in[3];
    const float* b_dec = (const float*)d_in[4];
    float* out = (float*)d_out;

    const int B = in_sizes[0];   // 2048
    const int H = in_sizes[2];   // 128  (kernel 2 assumes KH == 128)
    const int V = in_sizes[4];   // 100000

    float* hidden = (float*)d_ws;   // B*H*4 = 1 MB scratch

    // 1) gather hidden
    {
        int total = B * H;
        int blocks = (total + 255) / 256;
        sg_gather_hidden<<<blocks, 256, 0, stream>>>(idx, W_enc, b_enc, hidden, B, H, V);
    }
    // 2) WMMA GEMM + bias -> logits in d_out
    {
        dim3 grid((V + BLOCK_N - 1) / BLOCK_N, (B + BLOCK_M - 1) / BLOCK_M);
        sg_wmma_logits<<<grid, 256, 0, stream>>>(hidden, W_dec, b_dec, out, B, V);
    }
    // 3) row-wise log_softmax in place
    {
        sg_logsoftmax_rows<<<B, 256, 0, stream>>>(out, V);
    }
}